// TTEmbedding_48962627174843
// MI455X (gfx1250) — compile-verified
//
#include <hip/hip_runtime.h>

// TT-matrix embedding on gfx1250 (MI455X), wave32, fp32 WMMA 16x16x4.
//
// VOC_Q=(40,36,36), EMB_Q=(8,16,8), RANK=8.
// core1[40][8][8]      A[i1][e1][r1]
// core2[8][36][16][8]  B[r1][i2][e2][r2]   (r1 stride 4608, i2 stride 128)
// core3[8][36][8]      C[r2][i3][e3]       (r2 stride 288)
//
// Per wave: 2 tokens.
//  Stage1: D[16,128] = blockdiag(A_t0,A_t1) x [B_t0;B_t1]   (32 wmma)
//  Stage2: D[128,16] = [T_t0|T_t1] x blockdiag(C_t0,C_t1)   (32 wmma)
// Block-diagonal masking done with hoisted 0/1 multiplies (branchless, EXEC
// stays all-ones). LDS (8KB/wave) reshuffles between stages; output stored
// as coalesced b128 with PAD_IDX zeroing folded in.

typedef float v2f __attribute__((ext_vector_type(2)));
typedef float v4f __attribute__((ext_vector_type(4)));
typedef float v8f __attribute__((ext_vector_type(8)));

#define WAVES_PER_WG 4

__global__ __launch_bounds__(128) void tt_embedding_wmma_kernel(
    const int* __restrict__ x,
    const float* __restrict__ c1,
    const float* __restrict__ c2,
    const float* __restrict__ c3,
    float* __restrict__ out)
{
  __shared__ float smem[WAVES_PER_WG * 2048];

  const int lane = threadIdx.x & 31;
  const int w    = threadIdx.x >> 5;
  float* lT = &smem[w * 2048];              // [tok][128 rows][8] fp32

  const int lhi = lane >> 4;                // 0/1: K offset +0 / +2
  const int llo = lane & 15;                // long dim (M for A, N for B/D)

  const int p  = blockIdx.x * WAVES_PER_WG + w;   // token pair id
  const int t0 = p * 2;
  const int idx0 = x[t0];
  const int idx1 = x[t0 + 1];

  // idx -> (i1, i2, i3)
  const int i1_0 = idx0 / 1296; const int q0 = idx0 - i1_0 * 1296;
  const int i2_0 = q0 / 36;     const int i3_0 = q0 - i2_0 * 36;
  const int i1_1 = idx1 / 1296; const int q1 = idx1 - i1_1 * 1296;
  const int i2_1 = q1 / 36;     const int i3_1 = q1 - i2_1 * 36;

  // ---------------- Stage 1 hoisted A operand (nt-invariant) ----------------
  // A row m = llo: rows 0..7 = token0 e1, rows 8..15 = token1 e1.
  // K-step ks uses K = ks*4 + lhi*2 + {0,1}; (K&7) = lhi*2 + (ks&1)*4 + {0,1}.
  // Valid (nonzero) iff (K>=8) == (row>=8)  ->  mask by 0/1 multiply.
  const int aTok = (llo >= 8);
  const float* aRow = c1 + (aTok ? i1_1 : i1_0) * 64 + (llo & 7) * 8;
  const v2f aLo = *(const v2f*)&aRow[lhi * 2];       // r1 = lhi*2 + {0,1}
  const v2f aHi = *(const v2f*)&aRow[lhi * 2 + 4];   // r1 = lhi*2 + {4,5}
  const float mA0 = aTok ? 0.0f : 1.0f;              // K<8 block valid
  const float mA1 = 1.0f - mA0;                      // K>=8 block valid
  v2f aOp[4];
  aOp[0] = aLo * mA0;  aOp[1] = aHi * mA0;
  aOp[2] = aLo * mA1;  aOp[3] = aHi * mA1;

  // Stage-1 B row pointers (nt-invariant): K-step ks, vgpr v ->
  //   c2 + (K&7)*4608 + i2(tok of K)*128, tok = (ks>=2)
  const float* bRow[4][2];
#pragma unroll
  for (int ks = 0; ks < 4; ++ks) {
    const int k0  = ks * 4 + lhi * 2;
    const int i2B = (k0 >= 8) ? i2_1 : i2_0;
    bRow[ks][0] = c2 + (k0 & 7) * 4608 + i2B * 128;
    bRow[ks][1] = c2 + ((k0 + 1) & 7) * 4608 + i2B * 128;
  }

  // ---------------- Stage 1: T = blockdiag(A0,A1) x [B0;B1] ----------------
#pragma unroll
  for (int nt = 0; nt < 8; ++nt) {
    const int col = nt * 16 + llo;          // B column = e2*8 + r2 (0..127)
    v8f acc = {};
#pragma unroll
    for (int ks = 0; ks < 4; ++ks) {
      v2f b;
      b.x = bRow[ks][0][col];
      b.y = bRow[ks][1][col];
      acc = __builtin_amdgcn_wmma_f32_16x16x4_f32(
          false, aOp[ks], false, b, (short)0, acc, false, false);
    }
    // Scatter D tile -> lT[tok][e1*16+e2][r2]
#pragma unroll
    for (int v = 0; v < 8; ++v) {
      const int m1  = v + lhi * 8;          // D row (0..15)
      const int tok = (m1 >= 8);
      const int e1  = m1 & 7;
      lT[tok * 1024 + e1 * 128 + (col >> 3) * 8 + (col & 7)] = acc[v];
    }
  }

  __builtin_amdgcn_wave_barrier();          // ds stores precede stage-2 loads

  // ---------------- Stage 2 hoisted B operand (mt-invariant) ----------------
  const int cTok = (llo >= 8);              // D column half -> token / e3
  const int i3B  = cTok ? i3_1 : i3_0;
  const int e3   = llo & 7;
  const float* cBase = c3 + i3B * 8 + e3;   // index by r2 (stride 288)
  const float cv0 = cBase[(lhi * 2)     * 288];
  const float cv1 = cBase[(lhi * 2 + 1) * 288];
  const float cv4 = cBase[(lhi * 2 + 4) * 288];
  const float cv5 = cBase[(lhi * 2 + 5) * 288];
  const float mB0 = cTok ? 0.0f : 1.0f;     // K<8 (token0 rows of C2)
  const float mB1 = 1.0f - mB0;
  v2f bOp[4];
  bOp[0] = (v2f){cv0, cv1} * mB0;  bOp[1] = (v2f){cv4, cv5} * mB0;
  bOp[2] = (v2f){cv0, cv1} * mB1;  bOp[3] = (v2f){cv4, cv5} * mB1;

  // ---------------- Stage 2: out = [T0|T1] x blockdiag(C0,C1) --------------
#pragma unroll
  for (int mt = 0; mt < 8; ++mt) {
    const int mrow = mt * 16 + llo;         // A row = e1*16+e2 (0..127)
    v8f acc = {};
#pragma unroll
    for (int ks = 0; ks < 4; ++ks) {
      const int k0  = ks * 4 + lhi * 2;     // K = (tok<<3)|r2, r2 even
      const int aTk = (k0 >= 8);
      const v2f a   = *(const v2f*)&lT[aTk * 1024 + mrow * 8 + (k0 & 7)];
      acc = __builtin_amdgcn_wmma_f32_16x16x4_f32(
          false, a, false, bOp[ks], (short)0, acc, false, false);
    }
    __builtin_amdgcn_wave_barrier();        // tile reads precede in-place write
#pragma unroll
    for (int v = 0; v < 8; ++v) {
      const int m1 = mt * 16 + v + lhi * 8;
      lT[cTok * 1024 + m1 * 8 + e3] = acc[v];
    }
  }

  __builtin_amdgcn_wave_barrier();

  // ---------------- Coalesced b128 store with PAD_IDX zeroing ---------------
  const float s0 = (idx0 != 0) ? 1.0f : 0.0f;
  const float s1 = (idx1 != 0) ? 1.0f : 0.0f;
  float* outBase = out + (size_t)t0 * 1024;
#pragma unroll
  for (int it = 0; it < 16; ++it) {
    const int f   = it * 32 + lane;         // float4 index 0..511
    const int tok = f >> 8;
    const int off = (f & 255) * 4;
    v4f val = *(const v4f*)&lT[tok * 1024 + off];
    val *= (tok ? s1 : s0);
    *(v4f*)&outBase[tok * 1024 + off] = val;
  }
}

extern "C" void kernel_launch(void* const* d_in, const int* in_sizes, int n_in,
                              void* d_out, int out_size, void* d_ws, size_t ws_size,
                              hipStream_t stream) {
  (void)n_in; (void)out_size; (void)d_ws; (void)ws_size;
  const int*   x  = (const int*)d_in[0];    // [8192] token indices
  const float* c1 = (const float*)d_in[1];  // [40][8][8]
  const float* c2 = (const float*)d_in[2];  // [8][36][16][8]
  const float* c3 = (const float*)d_in[3];  // [8][36][8]
  float* out = (float*)d_out;               // [8192][1024]

  const int nTok   = in_sizes[0];                   // 8192
  const int blocks = nTok / (2 * WAVES_PER_WG);     // 2 tokens/wave, 4 waves/WG

  tt_embedding_wmma_kernel<<<blocks, 128, 0, stream>>>(x, c1, c2, c3, out);
}